// PathSampling_23373212024872
// MI455X (gfx1250) — compile-verified
//
#include <hip/hip_runtime.h>
#include <hip/hip_bf16.h>

// Problem constants from the reference
#define N_PATH 32   // paths per node  (== wave32 width — one lane per path)
#define L_PATH 8    // nodes per path
#define K_PATH 8    // top-k paths kept
#define WAVES_PER_BLOCK 8

// One wave32 per node: lane j owns candidate path j.
//  1) load 8 path ids (2x b128), mask by rand_len, gather centrality, sum -> score
//  2) in-wave total-order rank via 32 shuffles (score desc, index asc tie-break,
//     matching jax.lax.top_k's stable ordering)
//  3) lanes with rank < 8 write their masked path row (2x b128) and 7 edge ids
//     to output slot == rank.
__global__ __launch_bounds__(32 * WAVES_PER_BLOCK)
void path_topk_kernel(const int*   __restrict__ paths,      // [n, 32, 8]
                      const int*   __restrict__ edge_ids,   // [n, 32, 7]
                      const int*   __restrict__ rand_lens,  // [n, 32]
                      const float* __restrict__ centrality, // [N_GRAPH_NODES]
                      int*         __restrict__ paths_out,  // [n, 8, 8]
                      int*         __restrict__ edges_out,  // [n, 8, 7]
                      int n_node)
{
    const int lane = threadIdx.x & 31;
    const int node = blockIdx.x * WAVES_PER_BLOCK + (threadIdx.x >> 5);
    if (node >= n_node) return;

    const long long row = (long long)node * N_PATH + lane;   // candidate-path row

    // Early prefetch of this lane's edge row (28B). Only 8/32 lanes will consume
    // it, but 28B rows share 128B cachelines so most lines are touched anyway;
    // this hides the edge-row latency behind the gather + shuffle phase.
    // Emits global_prefetch_b8 on gfx1250.
    const int* erow = edge_ids + row * (L_PATH - 1);
    __builtin_prefetch(erow, 0, 1);

    // Path row: 8 ints, 32B-aligned -> two b128 loads.
    const int4* prow = (const int4*)(paths + row * L_PATH);
    int4 p0 = prow[0];
    int4 p1 = prow[1];
    const int rl = rand_lens[row];

    int ids[L_PATH] = {p0.x, p0.y, p0.z, p0.w, p1.x, p1.y, p1.z, p1.w};

    // Mask positions k > rl to -1; score = sum of centrality over kept ids.
    // Original ids are always in [0, n_graph), so "kept" <=> k <= rl.
    float score = 0.0f;
    #pragma unroll
    for (int k = 0; k < L_PATH; ++k) {
        if (k <= rl) {
            score += centrality[ids[k]];
        } else {
            ids[k] = -1;
        }
    }

    // Total-order rank across the wave: (score desc, lane asc). Ranks are a
    // permutation of 0..31, so exactly K_PATH lanes get rank < K_PATH.
    // __shfl on wave32 lowers to ds_bpermute_b32.
    int rank = 0;
    #pragma unroll
    for (int i = 0; i < N_PATH; ++i) {
        const float si = __shfl(score, i, 32);
        if (si > score || (si == score && i < lane)) ++rank;
    }

    if (rank < K_PATH) {
        const long long obase = (long long)node * K_PATH + rank;

        // Output path row: 32B-aligned -> two b128 stores of the masked ids.
        int4* pout = (int4*)(paths_out + obase * L_PATH);
        pout[0] = make_int4(ids[0], ids[1], ids[2], ids[3]);
        pout[1] = make_int4(ids[4], ids[5], ids[6], ids[7]);

        // Edge row: 28B, only 4B-aligned -> dword copies.
        int* eout = edges_out + obase * (L_PATH - 1);
        #pragma unroll
        for (int m = 0; m < L_PATH - 1; ++m) eout[m] = erow[m];
    }
}

extern "C" void kernel_launch(void* const* d_in, const int* in_sizes, int n_in,
                              void* d_out, int out_size, void* d_ws, size_t ws_size,
                              hipStream_t stream)
{
    (void)n_in; (void)out_size; (void)d_ws; (void)ws_size;

    const int*   paths      = (const int*)  d_in[0];  // [n, 32, 8]
    const int*   edge_ids   = (const int*)  d_in[1];  // [n, 32, 7]
    const int*   rand_lens  = (const int*)  d_in[2];  // [n, 32]
    const float* centrality = (const float*)d_in[3];  // [N_GRAPH_NODES]
    // d_in[4] is k_path (device scalar) == 8; compile-time constant here.

    const int n_node = in_sizes[2] / N_PATH;          // 100000

    int* out       = (int*)d_out;
    int* paths_out = out;                                        // [n, 8, 8]
    int* edges_out = out + (size_t)n_node * K_PATH * L_PATH;     // [n, 8, 7]

    const dim3 block(32 * WAVES_PER_BLOCK);
    const dim3 grid((n_node + WAVES_PER_BLOCK - 1) / WAVES_PER_BLOCK);
    hipLaunchKernelGGL(path_topk_kernel, grid, block, 0, stream,
                       paths, edge_ids, rand_lens, centrality,
                       paths_out, edges_out, n_node);
}